// WindowVarLoss_61057255080537
// MI455X (gfx1250) — compile-verified
//
#include <hip/hip_runtime.h>
#include <hip/hip_bf16.h>
#include <math.h>

// Problem constants (from setup_inputs)
#define NIMG 4
#define FCH  32      // feature channels of x
#define DCH  32      // channels of d
#define GH   480
#define GW   640
#define SH   120     // pooled h
#define SW   160     // pooled w
#define RR   20      // number of random maps
#define NPIX (NIMG*SH*SW)   // 76800
#define SPIX (SH*SW)        // 19200
// spatially padded interleaved activation layout: [n][PH][PW][32], zero border
#define PH   (SH + 2)       // 122
#define PW   (SW + 2)       // 162
#define FEATP ((size_t)NIMG * PH * PW * FCH)   // 2,529,792 floats

typedef __attribute__((ext_vector_type(2))) float v2f;
typedef __attribute__((ext_vector_type(8))) float v8f;

// ---------------------------------------------------------------------------
// 1) fused random_pooling + bilinear grid sample.
//    Pool: per 4x4 block argmax of rand*(gt>0.1) (float4 row reads), gather gt.
//    Sample: bilinear gather of x at the shifted index, written to the padded
//    channel-interleaved feat buffer (interior only; border stays zero).
// ---------------------------------------------------------------------------
__global__ void k_pool_sample(const float* __restrict__ gts,
                              const float* __restrict__ rand_r,
                              const float* __restrict__ x,
                              float* __restrict__ rg,
                              float* __restrict__ featP) {
  int t = blockIdx.x * blockDim.x + threadIdx.x;
  if (t >= NPIX) return;
  int pw = t % SW; int t2 = t / SW; int ph = t2 % SH; int nimg = t2 / SH;
  const float* g  = gts    + (size_t)nimg * GH * GW;
  const float* rr = rand_r + (size_t)nimg * GH * GW;
  float best = -1.0f; int bi = 0, bj = 0;
  #pragma unroll
  for (int i = 0; i < 4; ++i) {
    int row = ph * 4 + i;
    float4 g4 = *(const float4*)(g  + row * GW + pw * 4);
    float4 r4 = *(const float4*)(rr + row * GW + pw * 4);
    float gv[4] = {g4.x, g4.y, g4.z, g4.w};
    float rv[4] = {r4.x, r4.y, r4.z, r4.w};
    #pragma unroll
    for (int j = 0; j < 4; ++j) {
      float m = rv[j] * (gv[j] > 0.1f ? 1.0f : 0.0f);
      if (m > best) { best = m; bi = i; bj = j; }   // strict > keeps first max
    }
  }
  int row = ph * 4 + bi, col = pw * 4 + bj;
  int idx = row * GW + col;
  float gval = g[idx];
  rg[t] = (gval < 0.1f) ? 0.0f : gval;

  // ---- grid sample at shifted index ----
  idx += (GW / SW / 2 + (GH / SH / 2) * GW);   // +1282
  int scol = idx % GW; int srow = idx / GW;
  float ix = (float)scol * (float)(GW - 1) / (float)GW;
  float iy = (float)srow * (float)(GH - 1) / (float)GH;
  float x0f = floorf(ix), y0f = floorf(iy);
  int   x0 = (int)x0f,    y0 = (int)y0f;
  float wx = ix - x0f,    wy = iy - y0f;
  float w00 = (1.f - wx) * (1.f - wy);
  float w10 = wx * (1.f - wy);
  float w01 = (1.f - wx) * wy;
  float w11 = wx * wy;
  bool vx0 = (x0 >= 0) & (x0 <= GW - 1);
  bool vx1 = (x0 + 1 >= 0) & (x0 + 1 <= GW - 1);
  bool vy0 = (y0 >= 0) & (y0 <= GH - 1);
  bool vy1 = (y0 + 1 >= 0) & (y0 + 1 <= GH - 1);
  w00 *= (vx0 & vy0) ? 1.f : 0.f;
  w10 *= (vx1 & vy0) ? 1.f : 0.f;
  w01 *= (vx0 & vy1) ? 1.f : 0.f;
  w11 *= (vx1 & vy1) ? 1.f : 0.f;
  int xi0 = min(max(x0, 0), GW - 1),     xi1 = min(max(x0 + 1, 0), GW - 1);
  int yi0 = min(max(y0, 0), GH - 1),     yi1 = min(max(y0 + 1, 0), GH - 1);
  const float* xb = x + (size_t)nimg * FCH * GH * GW;
  size_t o00 = (size_t)yi0 * GW + xi0, o10 = (size_t)yi0 * GW + xi1;
  size_t o01 = (size_t)yi1 * GW + xi0, o11 = (size_t)yi1 * GW + xi1;
  float* outp = featP + ((((size_t)nimg * PH + (ph + 1)) * PW + (pw + 1)) << 5);
  #pragma unroll
  for (int c4 = 0; c4 < FCH; c4 += 4) {
    float4 v;
    float* vp = (float*)&v;
    #pragma unroll
    for (int j = 0; j < 4; ++j) {
      const float* f = xb + (size_t)(c4 + j) * GH * GW;
      vp[j] = f[o00] * w00 + f[o10] * w10 + f[o01] * w01 + f[o11] * w11;
    }
    *(float4*)(outp + c4) = v;
  }
}

// ---------------------------------------------------------------------------
// 2) 32->32 3x3 SAME conv via V_WMMA_F32_16X16X4_F32 implicit GEMM.
//    Weights staged in LDS (36KB). Activations come from the zero-padded
//    interleaved buffer: one base address per lane, all 72 A-fetches are
//    global_load_b64 at compile-time immediate offsets; no halo selects.
//    Fused sigmoid * d. One wave = 16px x 32oc tile; EXEC all-ones at WMMA.
// ---------------------------------------------------------------------------
__global__ void k_conv_att(const float* __restrict__ featP,
                           const float* __restrict__ Watt,
                           const float* __restrict__ batt,
                           const float* __restrict__ dmap,
                           float* __restrict__ tmpP) {
  __shared__ v2f BL[9 * 8 * 2 * 32];            // 36 KB of 320 KB WGP LDS
  for (int i = threadIdx.x; i < 9 * 8 * 2 * 32; i += 256) {
    int n   = i & 31;
    int hf  = (i >> 5) & 1;
    int cb  = (i >> 6) & 7;
    int tap = i >> 9;
    int cin = cb * 4 + 2 * hf;
    v2f w;
    w.x = Watt[n * (FCH * 9) + cin * 9 + tap];
    w.y = Watt[n * (FCH * 9) + (cin + 1) * 9 + tap];
    BL[i] = w;
  }
  __syncthreads();

  int gw   = blockIdx.x * 8 + (threadIdx.x >> 5);   // exact launch: gw < 4800
  int lane = threadIdx.x & 31;
  int xt = gw % (SW / 16); int t2 = gw / (SW / 16);
  int y = t2 % SH; int nimg = t2 / SH;
  int x0 = xt * 16;
  int hf  = lane >> 4;       // K half: {0,1} vs {2,3}
  int l15 = lane & 15;       // M for A, N for B
  v8f acc0 = {}; v8f acc1 = {};
  // per-lane base at padded center pixel (y+1, x0+l15+1)
  const v2f* fb = (const v2f*)(featP +
      ((((size_t)nimg * PH + (y + 1)) * PW + (x0 + l15 + 1)) << 5));
  #pragma unroll
  for (int tap = 0; tap < 9; ++tap) {
    const int dy = tap / 3 - 1, dx = tap % 3 - 1;
    const int doff = (dy * PW + dx) * 16;       // v2f units, immediate offset
    const v2f* bl = &BL[tap * 512];             // [cb][hf][n]
    #pragma unroll
    for (int cb = 0; cb < 8; ++cb) {
      v2f a  = fb[doff + cb * 2 + hf];          // global_load_b64, const offset
      v2f b0 = bl[(cb * 2 + hf) * 32 + l15];    // ds_load_b64 (pairs to 2addr)
      v2f b1 = bl[(cb * 2 + hf) * 32 + l15 + 16];
      acc0 = __builtin_amdgcn_wmma_f32_16x16x4_f32(false, a, false, b0,
                                                   (short)0, acc0, false, false);
      acc1 = __builtin_amdgcn_wmma_f32_16x16x4_f32(false, a, false, b1,
                                                   (short)0, acc1, false, false);
    }
  }
  int nout0 = l15, nout1 = l15 + 16;
  float bias0 = batt[nout0], bias1 = batt[nout1];
  int pixbase = y * SW + x0 + 8 * hf;             // 8 consecutive pixels per lane
  const float* d0 = dmap + (size_t)(nimg * DCH + nout0) * SPIX + pixbase;
  const float* d1 = dmap + (size_t)(nimg * DCH + nout1) * SPIX + pixbase;
  float* tb = tmpP +
      ((((size_t)nimg * PH + (y + 1)) * PW + (x0 + 8 * hf + 1)) << 5);
  #pragma unroll
  for (int v = 0; v < 8; ++v) {                   // D: M = v + 8*hf
    float a0 = 1.0f / (1.0f + __expf(-(acc0[v] + bias0)));
    float a1 = 1.0f / (1.0f + __expf(-(acc1[v] + bias1)));
    tb[(size_t)v * 32 + nout0] = a0 * d0[v];
    tb[(size_t)v * 32 + nout1] = a1 * d1[v];
  }
}

// ---------------------------------------------------------------------------
// 3) 32->4 3x3 SAME conv via WMMA; B zero-padded to N=16 in LDS (18KB).
//    Same padded-activation addressing as conv_att.
// ---------------------------------------------------------------------------
__global__ void k_conv_post(const float* __restrict__ tmpP,
                            const float* __restrict__ Wpost,
                            const float* __restrict__ bpost,
                            float* __restrict__ dsb) {
  __shared__ v2f BP[9 * 8 * 2 * 16];              // 18 KB
  for (int i = threadIdx.x; i < 9 * 8 * 2 * 16; i += 256) {
    int n   = i & 15;
    int hf  = (i >> 4) & 1;
    int cb  = (i >> 5) & 7;
    int tap = i >> 8;
    v2f w; w.x = 0.0f; w.y = 0.0f;
    if (n < 4) {
      int cin = cb * 4 + 2 * hf;
      w.x = Wpost[n * (DCH * 9) + cin * 9 + tap];
      w.y = Wpost[n * (DCH * 9) + (cin + 1) * 9 + tap];
    }
    BP[i] = w;
  }
  __syncthreads();

  int gw   = blockIdx.x * 8 + (threadIdx.x >> 5);
  int lane = threadIdx.x & 31;
  int xt = gw % (SW / 16); int t2 = gw / (SW / 16);
  int y = t2 % SH; int nimg = t2 / SH;
  int x0 = xt * 16;
  int hf  = lane >> 4;
  int l15 = lane & 15;
  v8f acc = {};
  const v2f* fb = (const v2f*)(tmpP +
      ((((size_t)nimg * PH + (y + 1)) * PW + (x0 + l15 + 1)) << 5));
  #pragma unroll
  for (int tap = 0; tap < 9; ++tap) {
    const int dy = tap / 3 - 1, dx = tap % 3 - 1;
    const int doff = (dy * PW + dx) * 16;
    const v2f* bl = &BP[tap * 256];
    #pragma unroll
    for (int cb = 0; cb < 8; ++cb) {
      v2f a = fb[doff + cb * 2 + hf];
      v2f b = bl[(cb * 2 + hf) * 16 + l15];
      acc = __builtin_amdgcn_wmma_f32_16x16x4_f32(false, a, false, b,
                                                  (short)0, acc, false, false);
    }
  }
  if (l15 < 4) {
    float bias = bpost[l15];
    #pragma unroll
    for (int v = 0; v < 8; ++v) {
      int m = v + 8 * hf;
      dsb[(size_t)(nimg * 4 + l15) * SPIX + y * SW + x0 + m] = acc[v] + bias;
    }
  }
}

// ---------------------------------------------------------------------------
// 4) directional log-gradients of rg + masked smooth-L1 vs ds; wave-reduce
//    then global atomic accumulate (sum_sl1, sum_mask) for this layer.
// ---------------------------------------------------------------------------
__global__ void k_loss(const float* __restrict__ rg,
                       const float* __restrict__ dsb,
                       float* __restrict__ accum2) {
  int t = blockIdx.x * blockDim.x + threadIdx.x;
  float s1 = 0.0f, s2 = 0.0f;
  if (t < NPIX) {
    int xw = t % SW; int t2 = t / SW; int yh = t2 % SH; int nimg = t2 / SH;
    const float* rb = rg + (size_t)nimg * SPIX;
    float c  = rb[yh * SW + xw];
    bool  mc = c > 0.1f;
    float lc = logf(c + 1e-6f);
    const int ooff[4][2] = {{0, 1}, {1, 1}, {1, 0}, {1, -1}};
    #pragma unroll
    for (int k = 0; k < 4; ++k) {
      int ny = yh + ooff[k][0], nx = xw + ooff[k][1];
      bool inb = (ny >= 0) & (ny < SH) & (nx >= 0) & (nx < SW);
      float nv = inb ? rb[ny * SW + nx] : 0.0f;    // zero padding
      if (mc && (nv > 0.1f)) {
        float g   = lc - logf(nv + 1e-6f);
        float dsv = dsb[(size_t)(nimg * 4 + k) * SPIX + yh * SW + xw];
        float a   = fabsf(dsv - g);
        float sl1 = (a < 0.01f) ? (0.5f * a * a / 0.01f) : (a - 0.005f);
        s1 += sl1; s2 += 1.0f;
      }
    }
  }
  #pragma unroll
  for (int off = 16; off > 0; off >>= 1) {
    s1 += __shfl_down(s1, off, 32);
    s2 += __shfl_down(s2, off, 32);
  }
  if ((threadIdx.x & 31) == 0) {
    atomicAdd(&accum2[0], s1);
    atomicAdd(&accum2[1], s2);
  }
}

__global__ void k_zero(float* __restrict__ p, long n) {
  long t = (long)blockIdx.x * blockDim.x + threadIdx.x;
  if (t < n) p[t] = 0.0f;
}

__global__ void k_final(const float* __restrict__ accum, float* __restrict__ out) {
  if (blockIdx.x == 0 && threadIdx.x == 0) {
    float total = 0.0f;
    for (int r = 0; r < RR; ++r) total += accum[2 * r] / accum[2 * r + 1];
    out[0] = total / (float)RR;
  }
}

// ---------------------------------------------------------------------------
extern "C" void kernel_launch(void* const* d_in, const int* in_sizes, int n_in,
                              void* d_out, int out_size, void* d_ws, size_t ws_size,
                              hipStream_t stream) {
  const float* x     = (const float*)d_in[0];
  const float* dmap  = (const float*)d_in[1];
  const float* gts   = (const float*)d_in[2];
  const float* rnd   = (const float*)d_in[3];
  const float* Watt  = (const float*)d_in[4];
  const float* batt  = (const float*)d_in[5];
  const float* Wpost = (const float*)d_in[6];
  const float* bpost = (const float*)d_in[7];
  float* out = (float*)d_out;
  float* ws  = (float*)d_ws;

  // workspace layout (floats): rg | featP | tmpP | dsb | accum[2*RR]
  float* rg    = ws;                                    // 76,800
  float* featP = ws + NPIX;                             // 2,529,792 (padded, interleaved)
  float* tmpP  = featP + FEATP;                         // 2,529,792
  float* dsb   = tmpP + FEATP;                          // 307,200
  float* accum = dsb + (size_t)NIMG * 4 * SPIX;         // 2*RR
  // zero featP..accum once: borders of featP/tmpP must start (and stay) zero,
  // accumulators must start zero. Interiors are fully rewritten every layer.
  long zcount = (long)(2 * FEATP) + (long)NIMG * 4 * SPIX + 2 * RR;
  k_zero<<<(int)((zcount + 255) / 256), 256, 0, stream>>>(featP, zcount);

  const int pixBlocks  = (NPIX + 255) / 256;            // 300
  const int convBlocks = (NIMG * SH * (SW / 16)) / 8;   // 600 (8 waves/block)

  for (int r = 0; r < RR; ++r) {
    const float* rr = rnd + (size_t)r * NIMG * GH * GW;
    k_pool_sample<<<pixBlocks,  256, 0, stream>>>(gts, rr, x, rg, featP);
    k_conv_att   <<<convBlocks, 256, 0, stream>>>(featP, Watt, batt, dmap, tmpP);
    k_conv_post  <<<convBlocks, 256, 0, stream>>>(tmpP, Wpost, bpost, dsb);
    k_loss       <<<pixBlocks,  256, 0, stream>>>(rg, dsb, accum + 2 * r);
  }
  k_final<<<1, 1, 0, stream>>>(accum, out);
}